// SageBackbone_45853070852695
// MI455X (gfx1250) — compile-verified
//
#include <hip/hip_runtime.h>

typedef __attribute__((ext_vector_type(2))) float v2f;
typedef __attribute__((ext_vector_type(8))) float v8f;

// ---------------------------------------------------------------------------
// Degree count: one thread per edge, atomic f32 add into deg[dst].
// ---------------------------------------------------------------------------
__global__ __launch_bounds__(256)
void deg_kernel(const long long* __restrict__ dst, float* __restrict__ deg, int n_edges) {
    int e = blockIdx.x * blockDim.x + threadIdx.x;
    if (e < n_edges) atomicAdd(&deg[dst[e]], 1.0f);
}

// ---------------------------------------------------------------------------
// Mean-aggregation scatter: thread t handles one float4 chunk of one edge.
// Coalesced float4 load of the source row, 4 scalar f32 atomics to dest row.
// ---------------------------------------------------------------------------
template<int D>
__global__ __launch_bounds__(256)
void scatter_kernel(const float* __restrict__ x,
                    const long long* __restrict__ src,
                    const long long* __restrict__ dst,
                    float* __restrict__ agg, int n_edges) {
    constexpr int PER_EDGE = D / 4;
    long long t = (long long)blockIdx.x * blockDim.x + threadIdx.x;
    long long e = t / PER_EDGE;
    int f4 = (int)(t % PER_EDGE);
    if (e >= n_edges) return;
    long long s = src[e];
    long long d = dst[e];
    float4 v = ((const float4*)(x + s * (long long)D))[f4];
    float* o = agg + d * (long long)D + (long long)f4 * 4;
    atomicAdd(o + 0, v.x);
    atomicAdd(o + 1, v.y);
    atomicAdd(o + 2, v.z);
    atomicAdd(o + 3, v.w);
}

// ---------------------------------------------------------------------------
// agg[node, :] *= 1 / max(deg[node], 1)
// ---------------------------------------------------------------------------
template<int D>
__global__ __launch_bounds__(256)
void scale_kernel(float* __restrict__ agg, const float* __restrict__ deg, int n_nodes) {
    constexpr int PER_NODE = D / 4;
    long long t = (long long)blockIdx.x * blockDim.x + threadIdx.x;
    long long node = t / PER_NODE;
    int f4 = (int)(t % PER_NODE);
    if (node >= n_nodes) return;
    float r = 1.0f / fmaxf(deg[node], 1.0f);
    float4* p = (float4*)(agg + node * (long long)D) + f4;
    float4 v = *p;
    v.x *= r; v.y *= r; v.z *= r; v.w *= r;
    *p = v;
}

// ---------------------------------------------------------------------------
// out[32-row tile, 256] = relu( A1 @ Wl^T + A2 @ Wr^T + bias )
//   A1, A2 : [N, K1] f32     Wl, Wr : [256, K1] f32 (row-major)
//
// Block = 256 threads = 8 wave32, covering 32 rows x 256 cols.
// The 32 x (2*K1) concatenated activation tile [A1 | A2] is staged in
// dynamic LDS. Wave w owns output N-tiles {2w, 2w+1} for BOTH 16-row
// M-tiles: a 2x2 grid of 16x16 accumulators. Each K-step of 4 loads
// 2 A fragments (LDS) + 2 B fragments (global/L2) and issues 4
// v_wmma_f32_16x16x4_f32 -> each fragment is reused twice.
//
// The K loop is split into the Wl half and the Wr half with hoisted,
// lane-dependent base pointers so unrolled loads fold the k offset into
// the instruction's immediate instead of per-iteration 64-bit adds.
//
// Fragment layouts per CDNA5 ISA (7.12.2):
//   A 16x4 f32 : lane holds row lane&15; VGPR i holds K = 2*(lane>>4)+i
//   B 4x16 f32 : lane holds col lane&15; VGPR i holds K = 2*(lane>>4)+i
//   C/D 16x16  : VGPR v holds row v + 8*(lane>>4), col = lane&15
// ---------------------------------------------------------------------------
template<int K1>
__global__ __launch_bounds__(256)
void sage_gemm_relu(const float* __restrict__ A1,
                    const float* __restrict__ A2,
                    const float* __restrict__ Wl,
                    const float* __restrict__ Wr,
                    const float* __restrict__ bias,
                    float* __restrict__ out, int n_nodes) {
    constexpr int K   = 2 * K1;
    constexpr int LDK = K + 4;            // row pad -> spread over LDS banks
    extern __shared__ float Atile[];      // 32 * LDK floats

    const int tid  = threadIdx.x;
    const int row0 = blockIdx.x * 32;
    const bool full = (row0 + 32) <= n_nodes;

    // Cooperative load of the 32 x K concatenated activation tile.
    if (full) {
        for (int idx = tid; idx < 32 * K; idx += 256) {
            int r = idx / K;
            int c = idx - r * K;
            long long gr = row0 + r;
            Atile[r * LDK + c] = (c < K1) ? A1[gr * K1 + c]
                                          : A2[gr * K1 + (c - K1)];
        }
    } else {
        for (int idx = tid; idx < 32 * K; idx += 256) {
            int r = idx / K;
            int c = idx - r * K;
            int gr = row0 + r;
            float v = 0.0f;
            if (gr < n_nodes)
                v = (c < K1) ? A1[(long long)gr * K1 + c]
                             : A2[(long long)gr * K1 + (c - K1)];
            Atile[r * LDK + c] = v;
        }
    }
    __syncthreads();

    const int wave = tid >> 5;
    const int lane = tid & 31;
    const int m    = lane & 15;           // A row / B col within a tile
    const int kq   = (lane >> 4) * 2;     // lane-half K offset: 0 or 2
    const int n0   = wave * 32;           // this wave's first output column

    v8f acc00 = {}, acc01 = {};           // M-tile 0, N-tiles 0/1
    v8f acc10 = {}, acc11 = {};           // M-tile 1, N-tiles 0/1

    // Hoisted lane-dependent base pointers; loop index folds into load imm.
    const float* ap0 = Atile + m * LDK + kq;            // A rows, M-tile 0
    const float* ap1 = Atile + (16 + m) * LDK + kq;     // A rows, M-tile 1
    const float* wl0 = Wl + (n0 + m) * K1 + kq;         // Wl, N-tile 0
    const float* wl1 = Wl + (n0 + 16 + m) * K1 + kq;    // Wl, N-tile 1
    const float* wr0 = Wr + (n0 + m) * K1 + kq;         // Wr, N-tile 0
    const float* wr1 = Wr + (n0 + 16 + m) * K1 + kq;    // Wr, N-tile 1

    // ---- Half 1: agg @ Wl^T ----
#pragma unroll 8
    for (int k = 0; k < K1; k += 4) {
        v2f a0, a1, b0, b1;
        a0.x = ap0[k];      a0.y = ap0[k + 1];
        a1.x = ap1[k];      a1.y = ap1[k + 1];
        b0.x = wl0[k];      b0.y = wl0[k + 1];
        b1.x = wl1[k];      b1.y = wl1[k + 1];
        acc00 = __builtin_amdgcn_wmma_f32_16x16x4_f32(false, a0, false, b0,
                                                      (short)0, acc00, false, false);
        acc01 = __builtin_amdgcn_wmma_f32_16x16x4_f32(false, a0, false, b1,
                                                      (short)0, acc01, false, false);
        acc10 = __builtin_amdgcn_wmma_f32_16x16x4_f32(false, a1, false, b0,
                                                      (short)0, acc10, false, false);
        acc11 = __builtin_amdgcn_wmma_f32_16x16x4_f32(false, a1, false, b1,
                                                      (short)0, acc11, false, false);
    }

    // ---- Half 2: self @ Wr^T (A columns K1..2K1 in the LDS tile) ----
#pragma unroll 8
    for (int k = 0; k < K1; k += 4) {
        v2f a0, a1, b0, b1;
        a0.x = ap0[K1 + k]; a0.y = ap0[K1 + k + 1];
        a1.x = ap1[K1 + k]; a1.y = ap1[K1 + k + 1];
        b0.x = wr0[k];      b0.y = wr0[k + 1];
        b1.x = wr1[k];      b1.y = wr1[k + 1];
        acc00 = __builtin_amdgcn_wmma_f32_16x16x4_f32(false, a0, false, b0,
                                                      (short)0, acc00, false, false);
        acc01 = __builtin_amdgcn_wmma_f32_16x16x4_f32(false, a0, false, b1,
                                                      (short)0, acc01, false, false);
        acc10 = __builtin_amdgcn_wmma_f32_16x16x4_f32(false, a1, false, b0,
                                                      (short)0, acc10, false, false);
        acc11 = __builtin_amdgcn_wmma_f32_16x16x4_f32(false, a1, false, b1,
                                                      (short)0, acc11, false, false);
    }

    // Epilogue: bias + relu + store.  D layout: VGPR v -> row v + 8*(lane>>4).
    const int mbase = (lane >> 4) * 8;
    const int ncol  = lane & 15;
    const float bia0 = bias[n0 + ncol];
    const float bia1 = bias[n0 + 16 + ncol];

    if (full) {
        float* o0 = out + (long long)(row0 + mbase) * 256 + n0 + ncol;
        float* o1 = o0 + 16LL * 256;
#pragma unroll
        for (int v = 0; v < 8; ++v) {
            o0[v * 256]      = fmaxf(acc00[v] + bia0, 0.0f);
            o0[v * 256 + 16] = fmaxf(acc01[v] + bia1, 0.0f);
            o1[v * 256]      = fmaxf(acc10[v] + bia0, 0.0f);
            o1[v * 256 + 16] = fmaxf(acc11[v] + bia1, 0.0f);
        }
    } else {
#pragma unroll
        for (int v = 0; v < 8; ++v) {
            long long r0 = row0 + mbase + v;
            long long r1 = r0 + 16;
            if (r0 < n_nodes) {
                out[r0 * 256 + n0 + ncol]      = fmaxf(acc00[v] + bia0, 0.0f);
                out[r0 * 256 + n0 + 16 + ncol] = fmaxf(acc01[v] + bia1, 0.0f);
            }
            if (r1 < n_nodes) {
                out[r1 * 256 + n0 + ncol]      = fmaxf(acc10[v] + bia0, 0.0f);
                out[r1 * 256 + n0 + 16 + ncol] = fmaxf(acc11[v] + bia1, 0.0f);
            }
        }
    }
}

// ---------------------------------------------------------------------------
// Host-side pipeline (graph-capture safe: only async ops on `stream`).
// ---------------------------------------------------------------------------
extern "C" void kernel_launch(void* const* d_in, const int* in_sizes, int n_in,
                              void* d_out, int out_size, void* d_ws, size_t ws_size,
                              hipStream_t stream) {
    const float*     x   = (const float*)d_in[0];
    const long long* ei  = (const long long*)d_in[1];   // int64 [2, E]
    const float*     Wl1 = (const float*)d_in[2];
    const float*     b1  = (const float*)d_in[3];
    const float*     Wr1 = (const float*)d_in[4];
    const float*     Wl2 = (const float*)d_in[5];
    const float*     b2  = (const float*)d_in[6];
    const float*     Wr2 = (const float*)d_in[7];

    const int N = in_sizes[0] / 128;   // 100000
    const int E = in_sizes[1] / 2;     // 640000
    const long long* src = ei;
    const long long* dst = ei + E;

    float* deg  = (float*)d_ws;                      // [N]
    float* agg1 = deg  + N;                          // [N,128]
    float* agg2 = agg1 + (size_t)N * 128;            // [N,256]
    float* h1   = agg2 + (size_t)N * 256;            // [N,256]
    float* out  = (float*)d_out;                     // [N,256]

    // Zero deg | agg1 | agg2 (contiguous) in one memset node.
    hipMemsetAsync(d_ws, 0, (size_t)N * (1 + 128 + 256) * sizeof(float), stream);

    // Degrees (shared by both layers).
    deg_kernel<<<(E + 255) / 256, 256, 0, stream>>>(dst, deg, E);

    // ---- Layer 1 ----
    {
        long long t = (long long)E * (128 / 4);
        scatter_kernel<128><<<(int)((t + 255) / 256), 256, 0, stream>>>(x, src, dst, agg1, E);
    }
    {
        long long t = (long long)N * (128 / 4);
        scale_kernel<128><<<(int)((t + 255) / 256), 256, 0, stream>>>(agg1, deg, N);
    }
    {
        size_t shmem = (size_t)32 * (2 * 128 + 4) * sizeof(float);   // 33,280 B
        sage_gemm_relu<128><<<(N + 31) / 32, 256, shmem, stream>>>(agg1, x, Wl1, Wr1, b1, h1, N);
    }

    // ---- Layer 2 ----
    {
        long long t = (long long)E * (256 / 4);
        scatter_kernel<256><<<(int)((t + 255) / 256), 256, 0, stream>>>(h1, src, dst, agg2, E);
    }
    {
        long long t = (long long)N * (256 / 4);
        scale_kernel<256><<<(int)((t + 255) / 256), 256, 0, stream>>>(agg2, deg, N);
    }
    {
        size_t shmem = (size_t)32 * (2 * 256 + 4) * sizeof(float);   // 66,048 B
        sage_gemm_relu<256><<<(N + 31) / 32, 256, shmem, stream>>>(agg2, h1, Wl2, Wr2, b2, out, N);
    }
}